// QNADE_3521873182819
// MI455X (gfx1250) — compile-verified
//
#include <hip/hip_runtime.h>
#include <math.h>

typedef __attribute__((ext_vector_type(2))) float v2f;
typedef __attribute__((ext_vector_type(4))) float v4f;
typedef __attribute__((ext_vector_type(8))) float v8f;

#if __has_builtin(__builtin_amdgcn_tanhf)
#define FAST_TANH(x) __builtin_amdgcn_tanhf(x)
#else
#define FAST_TANH(x) tanhf(x)
#endif

#define NSAMP   8192
#define DSITES  64
#define HDIM    4096
#define NT      16              // samples per block (one 16-row WMMA tile)
#define HS      (HDIM + 4)      // padded LDS row stride (floats); HS%4==0, bank-skew of 4
#define THREADS 256             // 8 waves (wave32)
#define NCHUNK  (HDIM / 32)     // 128 K-chunks of 4 per wave

// LDS layout in floats
#define ACC_OFF   0
#define W2R_OFF   (NT * HS)                   // 65600 ; [8 waves][2 half][3 mcol][128 i][2]
#define W2R_SZ    (8 * 2 * 3 * NCHUNK * 2)    // 12288 (48KB), slot mcol==2 is zeros
#define X_OFF     (W2R_OFF + W2R_SZ)          // 77888
#define OUTP_OFF  (X_OFF + NT * DSITES)       // 78912 ; 8 waves x 16 samples x 2
#define WAV_OFF   (OUTP_OFF + 8 * NT * 2)     // 79168
#define SMEM_FLOATS (WAV_OFF + NT)            // 79184 floats = 316736 B (< 320KB WGP LDS)

__global__ __launch_bounds__(THREADS)
void qnade_persistent_kernel(const float* __restrict__ xg,
                             const float* __restrict__ W1,
                             const float* __restrict__ b1,
                             const float* __restrict__ W2,
                             const float* __restrict__ b2,
                             float* __restrict__ outw) {
  extern __shared__ float smem[];
  float* accS = smem + ACC_OFF;   // [NT][HS]   running pre-activations
  float* W2r  = smem + W2R_OFF;   // lane-ordered B table (zeros in slot mcol==2)
  float* xS   = smem + X_OFF;     // [NT][DSITES]
  float* outP = smem + OUTP_OFF;  // [8 waves][NT][2] per-wave partial sums
  float* wavS = smem + WAV_OFF;   // [NT]

  const int tid  = threadIdx.x;
  const int lane = tid & 31;
  const int wave = tid >> 5;
  const int n0   = blockIdx.x * NT;

  // ---------- preload ----------
  // Lane-ordered, mask-baked B table:
  //   W2r[((wh*3 + m)*128 + i)*2 + j] = (m<2) ? W2[(w*512 + 4i + 2*half + j)*2 + m] : 0
  for (int idx = tid; idx < W2R_SZ; idx += THREADS) {
    const int j    = idx & 1;
    const int i    = (idx >> 1) & (NCHUNK - 1);
    const int m    = (idx >> 8) % 3;
    const int wh   = idx / (3 * 2 * NCHUNK);   // wave*2 + half
    const int w    = wh >> 1, half = wh & 1;
    float v = 0.0f;
    if (m < 2) v = W2[(size_t)(w * 512 + 4 * i + 2 * half + j) * 2 + m];
    W2r[idx] = v;
  }
  for (int i = tid; i < NT * DSITES; i += THREADS) {
    int n = i >> 6, dd = i & (DSITES - 1);
    xS[i] = xg[(size_t)(n0 + n) * DSITES + dd];
  }
  if (tid < NT) wavS[tid] = 1.0f;
  {
    // acc[n][h] = b1[h]  (mask==0 state before the first rank-1 update)
    const v4f* b1v4 = (const v4f*)b1;
    v4f bv[4];
#pragma unroll
    for (int j = 0; j < 4; ++j) bv[j] = b1v4[tid + j * THREADS];
    for (int n = 0; n < NT; ++n) {
      v4f* row = (v4f*)(accS + n * HS);   // n*HS is a multiple of 4 floats
#pragma unroll
      for (int j = 0; j < 4; ++j) row[tid + j * THREADS] = bv[j];
    }
  }
  const float b2_0 = b2[0];
  const float b2_1 = b2[1];

  // WMMA f32 16x16x4 lane roles (ISA layouts):
  //  A (16x4):  lane L -> row M = L&15 ; VGPR0/1 = K = (L<16 ? 0,1 : 2,3)
  //  B (4x16):  lane L -> col N = L&15 ; VGPR0/1 = K = (L<16 ? 0,1 : 2,3)
  //  C/D (16x16): VGPR v, lanes 0-15 -> M=v ; lanes 16-31 -> M=v+8 ; N = L&15
  const int   M     = lane & 15;
  const int   koff  = (lane < 16) ? 0 : 2;
  const int   baseM = (lane < 16) ? 0 : 8;
  const int   mcol  = (M < 2) ? M : 2;                 // slot 2 = zeros (mask baked in)
  const int   kbase = wave * (HDIM / 8) + koff;        // contiguous 512-float K range
  const v2f*  ap = (const v2f*)(accS + M * HS + kbase);
  const v2f*  bp = (const v2f*)(W2r + ((wave * 2 + (koff >> 1)) * 3 + mcol) * 2 * NCHUNK);

  // W1 row double-buffer (prefetch next row behind the WMMA loop)
  const v4f* W1v = (const v4f*)W1;
  v4f w1cur[4];
#pragma unroll
  for (int j = 0; j < 4; ++j) w1cur[j] = W1v[tid + j * THREADS];   // row 0

  for (int d = 0; d < DSITES; ++d) {
    __syncthreads();  // prev-step acc update complete; outP consumed

    // issue next W1 row load now; consumed after the WMMA loop
    const int dn = (d + 1 < DSITES) ? d + 1 : d;
    v4f w1nxt[4];
#pragma unroll
    for (int j = 0; j < 4; ++j)
      w1nxt[j] = W1v[(size_t)dn * (HDIM / 4) + tid + j * THREADS];

    // ---- out_partial = tanh(acc) @ W2 via V_WMMA_F32_16X16X4_F32 ----
    // Two independent accumulator chains so a WMMA from one chain can issue
    // while the other chain's tanh pair is still in its TRANS latency window.
    v8f cacc0 = {};
    v8f cacc1 = {};
#pragma unroll 4
    for (int i = 0; i < NCHUNK; i += 2) {
      v2f a0 = ap[2 * i];         // ds_load_b64 (pairs merge into 2addr_b64)
      v2f a1 = ap[2 * i + 2];
      v2f b0 = bp[i];             // ds_load_b64, contiguous per lane
      v2f b1 = bp[i + 1];
      a0.x = FAST_TANH(a0.x);
      a0.y = FAST_TANH(a0.y);
      a1.x = FAST_TANH(a1.x);
      a1.y = FAST_TANH(a1.y);
      cacc0 = __builtin_amdgcn_wmma_f32_16x16x4_f32(
          false, a0, false, b0, (short)0, cacc0, false, false);
      cacc1 = __builtin_amdgcn_wmma_f32_16x16x4_f32(
          false, a1, false, b1, (short)0, cacc1, false, false);
    }
    // deterministic per-wave partial dump (no float atomics)
    if (M < 2) {   // lanes 0,1,16,17 hold the two useful output columns
#pragma unroll
      for (int v = 0; v < 8; ++v)
        outP[(wave * NT + baseM + v) * 2 + M] = cacc0[v] + cacc1[v];
    }
    __syncthreads();  // partials visible; all acc reads for this step done

    // ---- finalize: fixed-order reduction, tanh, normalize, select ----
    if (tid < NT) {
      const int n = tid;
      float s0 = 0.0f, s1 = 0.0f;
#pragma unroll
      for (int w = 0; w < 8; ++w) {
        s0 += outP[(w * NT + n) * 2 + 0];
        s1 += outP[(w * NT + n) * 2 + 1];
      }
      float o0 = FAST_TANH(s0 + b2_0);
      float o1 = FAST_TANH(s1 + b2_1);
      float nr = fmaxf(sqrtf(o0 * o0 + o1 * o1), 1e-12f);
      float sel = (xS[n * DSITES + d] > 0.0f) ? o0 : o1;
      wavS[n] *= sel / nr;
    }

    // ---- rank-1 update: acc += x[:,d] (outer) W1[d,:]  (b128 DS ops) ----
    for (int n = 0; n < NT; ++n) {
      const float xd = xS[n * DSITES + d];
      v4f* row = (v4f*)(accS + n * HS);
#pragma unroll
      for (int j = 0; j < 4; ++j) {
        v4f v = row[tid + j * THREADS];
        v += xd * w1cur[j];
        row[tid + j * THREADS] = v;
      }
    }
#pragma unroll
    for (int j = 0; j < 4; ++j) w1cur[j] = w1nxt[j];
  }

  __syncthreads();
  if (tid < NT) outw[n0 + tid] = wavS[tid];
}

extern "C" void kernel_launch(void* const* d_in, const int* in_sizes, int n_in,
                              void* d_out, int out_size, void* d_ws, size_t ws_size,
                              hipStream_t stream) {
  (void)in_sizes; (void)n_in; (void)out_size; (void)d_ws; (void)ws_size;
  const float* x  = (const float*)d_in[0];
  const float* W1 = (const float*)d_in[1];
  const float* b1 = (const float*)d_in[2];
  const float* W2 = (const float*)d_in[3];
  const float* b2 = (const float*)d_in[4];
  float* out = (float*)d_out;

  const size_t shmem = (size_t)SMEM_FLOATS * sizeof(float);
  qnade_persistent_kernel<<<NSAMP / NT, THREADS, shmem, stream>>>(
      x, W1, b1, W2, b2, out);
}